// EGNNLayer_28613072126360
// MI455X (gfx1250) — compile-verified
//
#include <hip/hip_runtime.h>
#include <hip/hip_bf16.h>

// ---------------------------------------------------------------------------
// EGNN layer for MI455X (gfx1250), fp32 WMMA (V_WMMA_F32_16X16X4_F32).
// Edge kernel: one wave32 owns TWO 16-edge tiles -> each weight (B) fragment
// is reused for two WMMAs, halving L2 weight traffic per FLOP.
// Gathers go through GLOBAL_LOAD_ASYNC_TO_LDS_B128 when available (ASYNCcnt).
// ---------------------------------------------------------------------------

#define ND 128
#define ED 32
#define HD 256
#define KE 292            // 2*ND + 1 + ED = 289, padded to multiple of 4
#define EPB 32            // edges per block (two 16-row tiles per wave)

#define AS  292           // bufA row stride: 292 % 64 = 36 (gcd 4) -> 16 conflict-free rows
#define MS  260           // bufM row stride: 260 % 64 = 4  (gcd 4) -> 16 conflict-free rows
#define ATILE (16 * AS)
#define MTILE (16 * MS)

typedef float v2f __attribute__((ext_vector_type(2)));
typedef float v8f __attribute__((ext_vector_type(8)));

__device__ __forceinline__ float silu_f(float v) {
    return v / (1.0f + __expf(-v));
}

// ---------------- async global->LDS copy (CDNA5), guarded --------------------
#if defined(__gfx1250__) && __has_builtin(__builtin_amdgcn_global_load_async_to_lds_b128) && __has_builtin(__builtin_amdgcn_s_wait_asynccnt)
#define HAVE_ASYNC_LDS 1
#else
#define HAVE_ASYNC_LDS 0
#endif

#if HAVE_ASYNC_LDS
typedef int b128_t __attribute__((vector_size(16)));
typedef __attribute__((address_space(1))) b128_t* g128_t;
typedef __attribute__((address_space(3))) b128_t* l128_t;
#endif

__device__ __forceinline__ void copy16_g2l(const float* g, float* l) {
#if HAVE_ASYNC_LDS
    __builtin_amdgcn_global_load_async_to_lds_b128(
        (g128_t)(unsigned long long)(g),
        (l128_t)(unsigned int)(unsigned long long)(l),
        0, 0);
#else
    float4 v = *(const float4*)g;
    l[0] = v.x; l[1] = v.y; l[2] = v.z; l[3] = v.w;
#endif
}

__device__ __forceinline__ void g2l_wait() {
#if HAVE_ASYNC_LDS
    __builtin_amdgcn_s_wait_asynccnt(0);
#endif
}

// ---------------------------------------------------------------------------
// Dual-tile GEMM: D{0,1} = A{0,1}(16xK, LDS) * W(K x wn, global) + bias.
// B fragment loaded once per k-step, consumed by two WMMAs.
// WMMA f32 16x16x4 wave32 layouts:
//   A: a[r] = A[m][k0 + r + 2*(lane>>4)],  m = lane&15
//   B: b[r] = W[k0 + r + 2*(lane>>4)][n0 + (lane&15)]
//   D: vgpr r holds M = r + 8*(lane>>4), N = n0 + (lane&15)
// ---------------------------------------------------------------------------
__device__ __forceinline__ void wave_gemm2(const float* A0, const float* A1, int as,
                                           const float* __restrict__ W, int wn,
                                           const float* __restrict__ bias,
                                           float* O0, float* O1, int os,
                                           int K, bool dosilu, int lane) {
    const int nn = lane & 15;
    const int kh = (lane >> 4) * 2;
    const int mb = (lane >> 4) * 8;
    const float* a0 = &A0[(lane & 15) * as + kh];
    const float* a1 = &A1[(lane & 15) * as + kh];
    for (int n0 = 0; n0 < wn; n0 += 16) {
        v8f acc0 = {}, acc1 = {};
        const float* wc = &W[(long)kh * wn + n0 + nn];
        for (int k0 = 0; k0 < K; k0 += 4) {
            v2f b, p, q;
            b.x = wc[(long)k0 * wn];
            b.y = wc[(long)(k0 + 1) * wn];
            p.x = a0[k0]; p.y = a0[k0 + 1];
            q.x = a1[k0]; q.y = a1[k0 + 1];
            acc0 = __builtin_amdgcn_wmma_f32_16x16x4_f32(
                false, p, false, b, (short)0, acc0, false, false);
            acc1 = __builtin_amdgcn_wmma_f32_16x16x4_f32(
                false, q, false, b, (short)0, acc1, false, false);
        }
        float bs = bias[n0 + nn];
        for (int r = 0; r < 8; ++r) {
            float v0 = acc0[r] + bs;
            float v1 = acc1[r] + bs;
            if (dosilu) { v0 = silu_f(v0); v1 = silu_f(v1); }
            O0[(mb + r) * os + n0 + nn] = v0;
            O1[(mb + r) * os + n0 + nn] = v1;
        }
    }
}

// Single-tile GEMM (node kernel).
__device__ __forceinline__ void wave_gemm(const float* A0, int as,
                                          const float* __restrict__ W, int wn,
                                          const float* __restrict__ bias,
                                          float* O0, int os,
                                          int K, bool dosilu, int lane) {
    const int nn = lane & 15;
    const int kh = (lane >> 4) * 2;
    const int mb = (lane >> 4) * 8;
    const float* a0 = &A0[(lane & 15) * as + kh];
    for (int n0 = 0; n0 < wn; n0 += 16) {
        v8f acc = {};
        const float* wc = &W[(long)kh * wn + n0 + nn];
        for (int k0 = 0; k0 < K; k0 += 4) {
            v2f b, p;
            b.x = wc[(long)k0 * wn];
            b.y = wc[(long)(k0 + 1) * wn];
            p.x = a0[k0]; p.y = a0[k0 + 1];
            acc = __builtin_amdgcn_wmma_f32_16x16x4_f32(
                false, p, false, b, (short)0, acc, false, false);
        }
        float bs = bias[n0 + nn];
        for (int r = 0; r < 8; ++r) {
            float v = acc[r] + bs;
            if (dosilu) v = silu_f(v);
            O0[(mb + r) * os + n0 + nn] = v;
        }
    }
}

// ---------------------------------------------------------------------------
// Edge kernel: per 32-edge block (one wave): gather edge_in -> msg MLP
// (2 GEMMs) -> atomic segment-sum of m -> coord MLP GEMM + matvec ->
// atomic segment-sum of dx*coef.
// ---------------------------------------------------------------------------
__global__ __launch_bounds__(32) void egnn_edge_kernel(
    const float* __restrict__ h, const float* __restrict__ x,
    const int* __restrict__ ei, const float* __restrict__ ea,
    const float* __restrict__ W1p, const float* __restrict__ mb1,
    const float* __restrict__ mW2, const float* __restrict__ mb2,
    const float* __restrict__ cW1, const float* __restrict__ cb1,
    const float* __restrict__ cW2,
    float* __restrict__ m_acc, float* __restrict__ x_acc,
    int nE) {
    __shared__ float bufA[2 * ATILE];   // edge_in (2x16xKE) ; later m (2x16x256)
    __shared__ float bufM[2 * MTILE];   // m1 (2x16x256)     ; later coef-pre
    __shared__ float dxs[EPB * 3];
    __shared__ float dists[EPB];
    __shared__ int rows_s[EPB], cols_s[EPB], valid_s[EPB];

    const int lane = threadIdx.x;
    const long e0 = (long)blockIdx.x * EPB;

    {   // one edge per lane: indices, dx, dist_sq
        long e = e0 + lane;
        int v = (e < nE) ? 1 : 0;
        long es = v ? e : 0;
        int r = ei[es];
        int c = ei[(long)nE + es];
        rows_s[lane] = r; cols_s[lane] = c; valid_s[lane] = v;
        float dx = x[(long)r * 3 + 0] - x[(long)c * 3 + 0];
        float dy = x[(long)r * 3 + 1] - x[(long)c * 3 + 1];
        float dz = x[(long)r * 3 + 2] - x[(long)c * 3 + 2];
        dxs[lane * 3 + 0] = dx; dxs[lane * 3 + 1] = dy; dxs[lane * 3 + 2] = dz;
        dists[lane] = dx * dx + dy * dy + dz * dz + 1e-8f;
    }
    __syncthreads();

    // edge_in[:, 0:256] = [h[row] | h[col]] : async gather straight into LDS
    for (int t = lane; t < EPB * 64; t += 32) {
        int e = t >> 6, q = t & 63;
        const float* src = (q < 32) ? &h[(long)rows_s[e] * ND + q * 4]
                                    : &h[(long)cols_s[e] * ND + (q - 32) * 4];
        float* dst = &bufA[(e >> 4) * ATILE + (e & 15) * AS + q * 4];
        copy16_g2l(src, dst);
    }
    g2l_wait();
    // edge_in[:, 256:292] = [dist_sq | edge_attr | zero-pad]
    for (int t = lane; t < EPB * 36; t += 32) {
        int e = t / 36, j = t - e * 36;
        float v;
        if (j == 0)        v = dists[e];
        else if (j <= 32)  v = valid_s[e] ? ea[(e0 + e) * ED + (j - 1)] : 0.0f;
        else               v = 0.0f;
        bufA[(e >> 4) * ATILE + (e & 15) * AS + 256 + j] = v;
    }
    __syncthreads();

    // m1 = silu(edge_in @ mW1 + mb1)
    wave_gemm2(bufA, bufA + ATILE, AS, W1p, HD, mb1,
               bufM, bufM + MTILE, MS, KE, true, lane);
    __syncthreads();
    // m = silu(m1 @ mW2 + mb2)  -> bufA
    wave_gemm2(bufM, bufM + MTILE, MS, mW2, HD, mb2,
               bufA, bufA + ATILE, AS, HD, true, lane);
    __syncthreads();

    // m_i segment sum (fp32 atomics into L2-resident accumulator)
    for (int t = lane; t < EPB * HD; t += 32) {
        int e = t >> 8, n = t & 255;
        if (valid_s[e])
            atomicAdd(&m_acc[(long)rows_s[e] * HD + n],
                      bufA[(e >> 4) * ATILE + (e & 15) * AS + n]);
    }

    // coef_pre = silu(m @ cW1 + cb1)  -> bufM
    wave_gemm2(bufA, bufA + ATILE, AS, cW1, HD, cb1,
               bufM, bufM + MTILE, MS, HD, true, lane);
    __syncthreads();

    // coef = coef_pre @ cW2 (matvec, one edge per lane), then x_agg atomics
    {
        int e = lane;
        const float* mrow = &bufM[(e >> 4) * MTILE + (e & 15) * MS];
        float p = 0.0f;
        for (int n = 0; n < HD; ++n) p += mrow[n] * cW2[n];
        if (valid_s[e]) {
            long r3 = (long)rows_s[e] * 3;
            atomicAdd(&x_acc[r3 + 0], dxs[e * 3 + 0] * p);
            atomicAdd(&x_acc[r3 + 1], dxs[e * 3 + 1] * p);
            atomicAdd(&x_acc[r3 + 2], dxs[e * 3 + 2] * p);
        }
    }
}

// ---------------------------------------------------------------------------
// Node kernel: [h | m_i] -> node MLP (2 GEMMs) + residual + LayerNorm.
// ---------------------------------------------------------------------------
__global__ __launch_bounds__(32) void egnn_node_kernel(
    const float* __restrict__ h, const float* __restrict__ m_acc,
    const float* __restrict__ nW1, const float* __restrict__ nb1,
    const float* __restrict__ nW2, const float* __restrict__ nb2,
    const float* __restrict__ gamma, const float* __restrict__ beta,
    float* __restrict__ hout, int nN) {
    __shared__ float Asb[16 * 388];  // [h | m_i] 16x384 (388 % 64 = 4 -> conflict-free)
    __shared__ float H1[16 * MS];    // hidden 16x256
    __shared__ float O[16 * 132];    // out 16x128

    const int lane = threadIdx.x;
    const long i0 = (long)blockIdx.x * 16;

    for (int t = lane; t < 16 * 96; t += 32) {
        int i = t / 96, q = t - i * 96;
        long node = i0 + i; if (node >= nN) node = nN - 1;
        const float* src = (q < 32) ? &h[node * ND + q * 4]
                                    : &m_acc[node * HD + (q - 32) * 4];
        copy16_g2l(src, &Asb[i * 388 + q * 4]);
    }
    g2l_wait();
    __syncthreads();

    // hid = silu([h|m_i] @ nW1 + nb1)
    wave_gemm(Asb, 388, nW1, HD, nb1, H1, MS, ND + HD, true, lane);
    __syncthreads();
    // o = hid @ nW2 + nb2
    wave_gemm(H1, MS, nW2, ND, nb2, O, 132, HD, false, lane);
    __syncthreads();
    // residual: o += h
    for (int t = lane; t < 16 * ND; t += 32) {
        int i = t >> 7, n = t & 127;
        long node = i0 + i;
        if (node < nN) O[i * 132 + n] += h[node * ND + n];
    }
    __syncthreads();
    // LayerNorm over 128 features; two half-lanes per row
    {
        int i = lane & 15, half = lane >> 4;
        float s = 0.0f, ss = 0.0f;
        const float* row = &O[i * 132 + half * 64];
        for (int n = 0; n < 64; ++n) { float v = row[n]; s += v; ss += v * v; }
        s  += __shfl_xor(s, 16, 32);
        ss += __shfl_xor(ss, 16, 32);
        float mu = s * (1.0f / 128.0f);
        float var = ss * (1.0f / 128.0f) - mu * mu;
        float rstd = rsqrtf(var + 1e-5f);
        long node = i0 + i;
        if (node < nN) {
            for (int n = half * 64; n < half * 64 + 64; ++n)
                hout[node * ND + n] = (O[i * 132 + n] - mu) * rstd * gamma[n] + beta[n];
        }
    }
}

// ---------------------------------------------------------------------------
// Small helper kernels
// ---------------------------------------------------------------------------
__global__ void zero_f32_kernel(float* __restrict__ p, long n) {
    long i = (long)blockIdx.x * blockDim.x + threadIdx.x;
    long stride = (long)gridDim.x * blockDim.x;
    for (; i < n; i += stride) p[i] = 0.0f;
}

__global__ void pad_w1_kernel(const float* __restrict__ w, float* __restrict__ wp) {
    int i = blockIdx.x * blockDim.x + threadIdx.x;
    if (i < KE * HD) wp[i] = (i < 289 * HD) ? w[i] : 0.0f;
}

__global__ void x_out_kernel(const float* __restrict__ x, const float* __restrict__ xa,
                             float* __restrict__ xo, long n) {
    long i = (long)blockIdx.x * blockDim.x + threadIdx.x;
    if (i < n) xo[i] = x[i] + xa[i];
}

// ---------------------------------------------------------------------------
extern "C" void kernel_launch(void* const* d_in, const int* in_sizes, int n_in,
                              void* d_out, int out_size, void* d_ws, size_t ws_size,
                              hipStream_t stream) {
    const float* h    = (const float*)d_in[0];
    const float* x    = (const float*)d_in[1];
    const int*   ei   = (const int*)d_in[2];
    const float* ea   = (const float*)d_in[3];
    const float* mW1  = (const float*)d_in[4];
    const float* mb1  = (const float*)d_in[5];
    const float* mW2  = (const float*)d_in[6];
    const float* mb2  = (const float*)d_in[7];
    const float* nW1  = (const float*)d_in[8];
    const float* nb1  = (const float*)d_in[9];
    const float* nW2  = (const float*)d_in[10];
    const float* nb2  = (const float*)d_in[11];
    const float* cW1  = (const float*)d_in[12];
    const float* cb1  = (const float*)d_in[13];
    const float* cW2  = (const float*)d_in[14];
    const float* gm   = (const float*)d_in[15];
    const float* bt   = (const float*)d_in[16];

    const int nN = in_sizes[0] / ND;
    const int nE = in_sizes[3] / ED;

    // workspace layout: [W1p (292x256)] [m_acc (N x 256)] [x_acc (N x 3)]
    float* W1p   = (float*)d_ws;
    float* m_acc = W1p + (long)KE * HD;
    float* x_acc = m_acc + (long)nN * HD;

    float* hout = (float*)d_out;
    float* xout = hout + (long)nN * ND;

    long nzero = (long)nN * HD + (long)nN * 3;   // m_acc and x_acc are contiguous
    zero_f32_kernel<<<2048, 256, 0, stream>>>(m_acc, nzero);
    pad_w1_kernel<<<(KE * HD + 255) / 256, 256, 0, stream>>>(mW1, W1p);

    int eblocks = (nE + EPB - 1) / EPB;
    egnn_edge_kernel<<<eblocks, 32, 0, stream>>>(h, x, ei, ea, W1p, mb1, mW2, mb2,
                                                 cW1, cb1, cW2, m_acc, x_acc, nE);

    int nblocks = (nN + 15) / 16;
    egnn_node_kernel<<<nblocks, 32, 0, stream>>>(h, m_acc, nW1, nb1, nW2, nb2,
                                                 gm, bt, hout, nN);

    x_out_kernel<<<((long)3 * nN + 255) / 256, 256, 0, stream>>>(x, x_acc, xout, (long)3 * nN);
}